// ICAEncoder_66563403153483
// MI455X (gfx1250) — compile-verified
//
#include <hip/hip_runtime.h>
#include <hip/hip_bf16.h>
#include <math.h>
#include <stdint.h>

typedef __attribute__((ext_vector_type(2))) float v2f;
typedef __attribute__((ext_vector_type(8))) float v8f;

#define B_T 4096
#define N_T 64
#define D_T 256
#define OUT_T 512
#define HSTR 258   // == 2 mod 64: conflict-free row & column scans; rows 8B-aligned
#define MSTR 257
#define XSTR 513   // == 1 mod 64

static __device__ __forceinline__ void wait_async_zero() {
#if __has_builtin(__builtin_amdgcn_s_wait_asynccnt)
    __builtin_amdgcn_s_wait_asynccnt(0);
#else
    asm volatile("s_wait_asynccnt 0" ::: "memory");
#endif
}

// ---------------------------------------------------------------------------
// Kernel 1: per-batch soft clustering. One block per b. h[b] staged into LDS
// once via CDNA5 async global->LDS DMA (bypasses VGPRs, tracked by ASYNCcnt).
// ---------------------------------------------------------------------------
__global__ __launch_bounds__(256) void ica_mu_kernel(
    const float* __restrict__ h, const float* __restrict__ log_tau,
    float* __restrict__ out_alpha, float* __restrict__ out_mu)
{
    extern __shared__ float sm[];
    float* h_s     = sm;                       // 64*258
    float* psum    = h_s  + N_T * HSTR;        // 4*64
    float* rn      = psum + 256;               // 64
    float* mu_s    = rn   + 64;                // 256
    float* simb    = mu_s + 256;               // 64
    float* ebuf    = simb + 64;                // 64
    float* alpha_s = ebuf + 64;                // 64
    float* red     = alpha_s + 64;             // 8
    float* scal    = red + 8;                  // 4

    const int t = threadIdx.x;
    const int b = blockIdx.x;
    const float* hb = h + (size_t)b * (N_T * D_T);

    // Stage h[b] (64x256) into padded LDS via async B64 DMA (8B chunks,
    // 8B-aligned in both global and padded LDS; wave covers 256B/issue).
    #pragma unroll
    for (int i = 0; i < 32; ++i) {
        int g = i * 512 + t * 2;               // float index, even
        int row = g >> 8, col = g & 255;
        unsigned lds_addr = (unsigned)(uintptr_t)(&h_s[row * HSTR + col]);
        unsigned long long gaddr = (unsigned long long)(uintptr_t)(hb + g);
        asm volatile("global_load_async_to_lds_b64 %0, %1, off"
                     :: "v"(lds_addr), "v"(gaddr) : "memory");
    }
    wait_async_zero();
    __syncthreads();

    // Row L2 norms: 4 threads per row, interleaved d = c + 4k.
    {
        int j = t & 63, c = t >> 6;
        float s = 0.f;
        #pragma unroll 8
        for (int k = 0; k < 64; ++k) {
            float x = h_s[j * HSTR + c + 4 * k];
            s += x * x;
        }
        psum[c * 64 + j] = s;
    }
    __syncthreads();
    if (t < 64) {
        float s = psum[t] + psum[64 + t] + psum[128 + t] + psum[192 + t];
        rn[t] = fmaxf(sqrtf(s), 1e-12f);
    }

    // mu init = column mean; thread t owns mu[t].
    float mu_d = 0.f;
    #pragma unroll 8
    for (int j = 0; j < 64; ++j) mu_d += h_s[j * HSTR + t];
    mu_d *= (1.f / 64.f);

    const float tau = fminf(fmaxf(expf(log_tau[0]), 0.01f), 2.0f);

    for (int it = 0; it < 3; ++it) {
        // publish mu, reduce ||mu||^2 (wave shuffle + cross-wave LDS)
        mu_s[t] = mu_d;
        float ss = mu_d * mu_d;
        #pragma unroll
        for (int off = 16; off > 0; off >>= 1) ss += __shfl_xor(ss, off, 32);
        if ((t & 31) == 0) red[t >> 5] = ss;
        __syncthreads();
        if (t == 0) {
            float s = 0.f;
            #pragma unroll
            for (int w = 0; w < 8; ++w) s += red[w];
            scal[0] = fmaxf(sqrtf(s), 1e-12f);
        }
        __syncthreads();
        const float norm = scal[0];

        // dots h_j . mu : 4 threads per row
        {
            int j = t & 63, c = t >> 6;
            float s = 0.f;
            #pragma unroll 8
            for (int k = 0; k < 64; ++k) {
                int d = c + 4 * k;
                s += h_s[j * HSTR + d] * mu_s[d];
            }
            psum[c * 64 + j] = s;
        }
        __syncthreads();
        if (t < 64) {
            float dot = psum[t] + psum[64 + t] + psum[128 + t] + psum[192 + t];
            simb[t] = dot / (rn[t] * norm * tau);
        }
        __syncthreads();
        // softmax over 64 (broadcast LDS reads; all threads run the loops)
        {
            float m = -INFINITY;
            #pragma unroll 8
            for (int i = 0; i < 64; ++i) m = fmaxf(m, simb[i]);
            if (t < 64) ebuf[t] = expf(simb[t] - m);
        }
        __syncthreads();
        {
            float s = 0.f;
            #pragma unroll 8
            for (int i = 0; i < 64; ++i) s += ebuf[i];
            if (t < 64) alpha_s[t] = ebuf[t] / s;
        }
        __syncthreads();
        // mu = sum_j alpha_j * h_j (thread t owns column t)
        float nm = 0.f;
        #pragma unroll 8
        for (int j = 0; j < 64; ++j) nm += alpha_s[j] * h_s[j * HSTR + t];
        mu_d = nm;
        __syncthreads();
    }

    if (t < 64) out_alpha[(size_t)b * 64 + t] = alpha_s[t];
    out_mu[(size_t)b * 256 + t] = mu_d;
}

// ---------------------------------------------------------------------------
// Kernel 2: fused MLP: (mu@w1^T+b1) -> LN -> GELU(exact) -> (@w2^T+b2) -> LN
// 16 rows per block, 8 waves, fp32 WMMA 16x16x4 (4 N-tiles per wave).
// ---------------------------------------------------------------------------
__global__ __launch_bounds__(256) void ica_mlp_kernel(
    const float* __restrict__ mu,
    const float* __restrict__ w1, const float* __restrict__ b1,
    const float* __restrict__ g1, const float* __restrict__ be1,
    const float* __restrict__ w2, const float* __restrict__ b2,
    const float* __restrict__ g2, const float* __restrict__ be2,
    float* __restrict__ theta)
{
    __shared__ float mu_t[16 * MSTR];  // 16x256 padded
    __shared__ float x_t[16 * XSTR];   // 16x512 padded (x1, then reused for x2)

    const int t    = threadIdx.x;
    const int lane = t & 31;
    const int wv   = t >> 5;
    const int lm   = lane & 15;   // M (A/C) or N (B/C) index in tile
    const int hi   = lane >> 4;   // K-pair / M-half selector
    const int row0 = blockIdx.x * 16;

    // Stage 16x256 mu tile.
    #pragma unroll
    for (int i = 0; i < 4; ++i) {
        int g = i * 1024 + t * 4;
        int r = g >> 8, c = g & 255;
        float4 v = *(const float4*)(mu + (size_t)(row0 + r) * 256 + c);
        float* dst = &mu_t[r * MSTR + c];
        dst[0] = v.x; dst[1] = v.y; dst[2] = v.z; dst[3] = v.w;
    }
    __syncthreads();

    // ---- GEMM1: x1 = mu @ w1^T  (M=16, K=256, N=512) ----
    v8f acc[4] = {};
    for (int ks = 0; ks < 64; ++ks) {
        int kb = 4 * ks + 2 * hi;
        v2f a;
        a.x = mu_t[lm * MSTR + kb];
        a.y = mu_t[lm * MSTR + kb + 1];
        #pragma unroll
        for (int tl = 0; tl < 4; ++tl) {
            int n = (wv * 4 + tl) * 16 + lm;
            const float* wp = w1 + (size_t)n * 256 + kb;
            v2f bf; bf.x = wp[0]; bf.y = wp[1];
            acc[tl] = __builtin_amdgcn_wmma_f32_16x16x4_f32(
                false, a, false, bf, (short)0, acc[tl], false, false);
        }
    }
    // Stage x1 + bias into LDS.
    #pragma unroll
    for (int tl = 0; tl < 4; ++tl) {
        int col = (wv * 4 + tl) * 16 + lm;
        float bb = b1[col];
        #pragma unroll
        for (int r = 0; r < 8; ++r) {
            int M = r + 8 * hi;
            x_t[M * XSTR + col] = acc[tl][r] + bb;
        }
    }
    __syncthreads();

    // ---- LN1 + exact GELU (each wave handles 2 rows; shuffle reductions) ----
    #pragma unroll
    for (int rr = 0; rr < 2; ++rr) {
        int R = wv * 2 + rr;
        float vals[16];
        float s = 0.f, sq = 0.f;
        #pragma unroll
        for (int k = 0; k < 16; ++k) {
            float x = x_t[R * XSTR + lane + 32 * k];
            vals[k] = x; s += x; sq += x * x;
        }
        #pragma unroll
        for (int off = 16; off > 0; off >>= 1) {
            s  += __shfl_xor(s,  off, 32);
            sq += __shfl_xor(sq, off, 32);
        }
        float m    = s * (1.f / 512.f);
        float var  = sq * (1.f / 512.f) - m * m;
        float rstd = rsqrtf(var + 1e-5f);
        #pragma unroll
        for (int k = 0; k < 16; ++k) {
            int col = lane + 32 * k;
            float y = (vals[k] - m) * rstd * g1[col] + be1[col];
            y = 0.5f * y * (1.f + erff(y * 0.70710678118654752f));
            x_t[R * XSTR + col] = y;
        }
    }
    __syncthreads();

    // ---- GEMM2: x2 = x1 @ w2^T  (M=16, K=512, N=512) ----
    v8f acc2[4] = {};
    for (int ks = 0; ks < 128; ++ks) {
        int kb = 4 * ks + 2 * hi;
        v2f a;
        a.x = x_t[lm * XSTR + kb];
        a.y = x_t[lm * XSTR + kb + 1];
        #pragma unroll
        for (int tl = 0; tl < 4; ++tl) {
            int n = (wv * 4 + tl) * 16 + lm;
            const float* wp = w2 + (size_t)n * 512 + kb;
            v2f bf; bf.x = wp[0]; bf.y = wp[1];
            acc2[tl] = __builtin_amdgcn_wmma_f32_16x16x4_f32(
                false, a, false, bf, (short)0, acc2[tl], false, false);
        }
    }
    __syncthreads();  // all x_t reads done before reuse

    #pragma unroll
    for (int tl = 0; tl < 4; ++tl) {
        int col = (wv * 4 + tl) * 16 + lm;
        float bb = b2[col];
        #pragma unroll
        for (int r = 0; r < 8; ++r) {
            int M = r + 8 * hi;
            x_t[M * XSTR + col] = acc2[tl][r] + bb;
        }
    }
    __syncthreads();

    // ---- LN2 -> theta ----
    #pragma unroll
    for (int rr = 0; rr < 2; ++rr) {
        int R = wv * 2 + rr;
        float vals[16];
        float s = 0.f, sq = 0.f;
        #pragma unroll
        for (int k = 0; k < 16; ++k) {
            float x = x_t[R * XSTR + lane + 32 * k];
            vals[k] = x; s += x; sq += x * x;
        }
        #pragma unroll
        for (int off = 16; off > 0; off >>= 1) {
            s  += __shfl_xor(s,  off, 32);
            sq += __shfl_xor(sq, off, 32);
        }
        float m    = s * (1.f / 512.f);
        float var  = sq * (1.f / 512.f) - m * m;
        float rstd = rsqrtf(var + 1e-5f);
        float* out = theta + (size_t)(row0 + R) * 512;
        #pragma unroll
        for (int k = 0; k < 16; ++k) {
            int col = lane + 32 * k;
            out[col] = (vals[k] - m) * rstd * g2[col] + be2[col];
        }
    }
}

// ---------------------------------------------------------------------------
extern "C" void kernel_launch(void* const* d_in, const int* in_sizes, int n_in,
                              void* d_out, int out_size, void* d_ws, size_t ws_size,
                              hipStream_t stream) {
    const float* h       = (const float*)d_in[0];
    const float* log_tau = (const float*)d_in[1];
    const float* w1      = (const float*)d_in[2];
    const float* b1      = (const float*)d_in[3];
    const float* g1      = (const float*)d_in[4];
    const float* be1     = (const float*)d_in[5];
    const float* w2      = (const float*)d_in[6];
    const float* b2      = (const float*)d_in[7];
    const float* g2      = (const float*)d_in[8];
    const float* be2     = (const float*)d_in[9];

    float* out   = (float*)d_out;
    float* theta = out;                                   // [4096, 512]
    float* alpha = out + (size_t)B_T * OUT_T;             // [4096, 64]
    float* mu    = alpha + (size_t)B_T * N_T;             // [4096, 256]

    size_t shmem1 = (size_t)(N_T * HSTR + 256 + 64 + 256 + 64 + 64 + 64 + 8 + 4)
                    * sizeof(float);

    ica_mu_kernel<<<dim3(B_T), dim3(256), shmem1, stream>>>(h, log_tau, alpha, mu);
    ica_mlp_kernel<<<dim3(B_T / 16), dim3(256), 0, stream>>>(
        mu, w1, b1, g1, be1, w2, b2, g2, be2, theta);
}